// ModelRCNN_13065290514475
// MI455X (gfx1250) — compile-verified
//
#include <hip/hip_runtime.h>
#include <stdint.h>

#define EPS_F       1e-5f
#define IOU_SCALE_F 10000.0f
#define NEG_TH      3000.0f
#define POS_TH      5000.0f
#define NUM_POS     128.0f
#define B_SZ        8
#define N_AC        147456            // 128*128*9
#define N_GT        64
#define TOTAL       (B_SZ * N_AC)     // 1179648
#define HALF_TOTAL  (TOTAL / 2)       // 589824

typedef float v2f __attribute__((ext_vector_type(2)));
typedef float v8f __attribute__((ext_vector_type(8)));

// ---------------- Threefry-2x32-20 (matches jax._src.prng) ----------------
__device__ __forceinline__ uint32_t tf_word(uint32_t k0, uint32_t k1,
                                            uint32_t c0, uint32_t c1, int pick) {
  uint32_t ks2 = k0 ^ k1 ^ 0x1BD11BDAu;
  uint32_t x0 = c0 + k0, x1 = c1 + k1;
#define TFR(rr) { x0 += x1; x1 = (x1 << rr) | (x1 >> (32 - rr)); x1 ^= x0; }
  TFR(13) TFR(15) TFR(26) TFR(6)   x0 += k1;  x1 += ks2 + 1u;
  TFR(17) TFR(29) TFR(16) TFR(24)  x0 += ks2; x1 += k0  + 2u;
  TFR(13) TFR(15) TFR(26) TFR(6)   x0 += k0;  x1 += k1  + 3u;
  TFR(17) TFR(29) TFR(16) TFR(24)  x0 += k1;  x1 += ks2 + 4u;
  TFR(13) TFR(15) TFR(26) TFR(6)   x0 += ks2; x1 += k0  + 5u;
#undef TFR
  return pick ? x1 : x0;
}

// Async global->LDS staging of one float4 (CDNA5 ASYNCcnt path).
__device__ __forceinline__ void async_stage_b128(float4* lds_dst, const float4* g_src) {
  const uint32_t ldsOff = (uint32_t)(uintptr_t)lds_dst;   // addrspace(3) byte offset
  const uint64_t gaddr  = (uint64_t)(uintptr_t)g_src;
  asm volatile("global_load_async_to_lds_b128 %0, %1, off"
               :: "v"(ldsOff), "v"(gaddr) : "memory");
}
__device__ __forceinline__ void async_wait0() {
  asm volatile("s_wait_asynccnt 0" ::: "memory");
}

// ---------------- Kernel A: zero per-batch positive counts ----------------
__global__ void rpn_init_counts(int* __restrict__ cnt) {
  if (threadIdx.x < B_SZ) cnt[threadIdx.x] = 0;
}

// ---------------- Kernel B: IOU argmax, proposals, flags, counts ----------
__global__ void __launch_bounds__(256)
rpn_assign(const float* __restrict__ anchors, const float* __restrict__ gt,
           const float* __restrict__ deltas, float* __restrict__ out,
           int* __restrict__ cnt) {
  __shared__ float4 sG4[N_GT];
  __shared__ float  sArea[N_GT];
  __shared__ int    sCnt;
  const int b = blockIdx.y;
  const int i = blockIdx.x * 256 + threadIdx.x;
  const size_t e = (size_t)b * N_AC + i;

  if (threadIdx.x == 0) sCnt = 0;
  if (threadIdx.x < N_GT)
    async_stage_b128(&sG4[threadIdx.x], &((const float4*)gt)[b * N_GT + threadIdx.x]);
  __builtin_prefetch(&deltas[e * 4], 0, 3);   // global_prefetch_b8
  async_wait0();                              // issuing waves drain ASYNCcnt
  if (threadIdx.x < N_GT) {
    float4 g = sG4[threadIdx.x];
    sArea[threadIdx.x] = fmaxf(g.z - g.x, 0.f) * fmaxf(g.w - g.y, 0.f);
  }
  __syncthreads();

  const float4 a = ((const float4*)anchors)[i];     // [y0,x0,y1,x1]
  const float areaA = fmaxf(a.z - a.x, 0.f) * fmaxf(a.w - a.y, 0.f);

  float best = -1.f; int bidx = 0;
  for (int g = 0; g < N_GT; ++g) {
    const float4 gb = sG4[g];                       // broadcast ds_load_b128
    const float yi = fmaxf(a.x, gb.x);
    const float xi = fmaxf(a.y, gb.y);
    const float ya = fminf(a.z, gb.z);
    const float xa = fminf(a.w, gb.w);
    const float inter = fmaxf(ya - yi, 0.f) * fmaxf(xa - xi, 0.f);
    const float uni = areaA + sArea[g] - inter;
    const float iou = (IOU_SCALE_F * inter) / (uni + EPS_F);
    if (iou > best) { best = iou; bidx = g; }       // first-max semantics
  }
  const bool pos = best > POS_TH;
  const bool neg = best < NEG_TH;
  float coor_sum;
  if (pos) { float4 gb = sG4[bidx]; coor_sum = gb.x + gb.y + gb.z + gb.w; }
  else     { coor_sum = neg ? -8.f : -4.f; }
  const bool basePos = coor_sum > -4.f;
  if (basePos) atomicAdd(&sCnt, 1);

  // proposals = delta2bbox(anchors, deltas)
  const float4 dl = ((const float4*)deltas)[e];
  const float wb = a.w - a.y, hb = a.z - a.x;
  const float xc = a.y + 0.5f * wb + wb * dl.x;
  const float yc = a.x + 0.5f * hb + hb * dl.y;
  const float w_ = wb * expf(dl.z);
  const float h_ = hb * expf(dl.w);
  ((float4*)out)[e * 2 + 0] =
      make_float4(yc - 0.5f * h_, xc - 0.5f * w_, yc + 0.5f * h_, xc + 0.5f * w_);

  // stash {idx, pos, basePos} for the finalize pass (overwritten there)
  const uint32_t meta = (uint32_t)bidx | ((uint32_t)pos << 8) | ((uint32_t)basePos << 9);
  out[e * 8 + 4] = __uint_as_float(meta);

  __syncthreads();
  if (threadIdx.x == 0) atomicAdd(&cnt[b], sCnt);
}

// ------- Kernel C: WMMA one-hot gather of gt_best + masked bbox2delta -----
// gt_best = onehot(best_idx) @ gt_boxes, exact in f32, via V_WMMA_F32_16X16X4_F32.
// Chain 1: GT coords in A rows 0-3  -> D rows 0-3  (lanes 0-15,  anchors lane 0..15)
// Chain 2: GT coords in A rows 8-11 -> D rows 8-11 (lanes 16-31, anchors lane 16..31)
// Both chains accumulate into one v8f; every lane ends with its coords in acc[0..3].
__global__ void __launch_bounds__(256)
rpn_finalize(const float* __restrict__ anchors, const float* __restrict__ gt,
             float* __restrict__ out, const int* __restrict__ cnt,
             uint32_t kp0, uint32_t kp1) {
  __shared__ float4 sG4[N_GT];
  const int b = blockIdx.y;
  const int i = blockIdx.x * 256 + threadIdx.x;
  const size_t e = (size_t)b * N_AC + i;

  if (threadIdx.x < N_GT)
    async_stage_b128(&sG4[threadIdx.x], &((const float4*)gt)[b * N_GT + threadIdx.x]);
  async_wait0();
  __syncthreads();
  __builtin_amdgcn_s_wait_tensorcnt(0);
  const float* sG = (const float*)sG4;

  const uint32_t meta = __float_as_uint(out[e * 8 + 4]);
  const int  myIdx   = (int)(meta & 0xFFu);
  const bool pos     = (meta >> 8) & 1u;
  const bool basePos = (meta >> 9) & 1u;
  const int  partner = __shfl_xor(myIdx, 16, 32);

  const int  lane = threadIdx.x & 31;
  const int  m    = lane & 15;
  const bool lo   = lane < 16;
  const int  colIdx1 = lo ? myIdx   : partner;  // B columns = anchors 0..15 of tile
  const int  colIdx2 = lo ? partner : myIdx;    // B columns = anchors 16..31 of tile
  const bool row1 = (m < 4);
  const bool row2 = (m >= 8) && (m < 12);
  const int  mc   = m & 3;

  v8f acc = {0.f, 0.f, 0.f, 0.f, 0.f, 0.f, 0.f, 0.f};
  for (int kb = 0; kb < N_GT; kb += 4) {
    // A 16x4 f32 layout: VGPR0 = K0|K2, VGPR1 = K1|K3 (lanes 0-15 | 16-31)
    const int   kA  = kb + (lo ? 0 : 2);
    const float gv0 = sG[kA * 4 + mc];
    const float gv1 = sG[(kA + 1) * 4 + mc];
    v2f A1 = { row1 ? gv0 : 0.f, row1 ? gv1 : 0.f };
    v2f A2 = { row2 ? gv0 : 0.f, row2 ? gv1 : 0.f };
    v2f B1 = { (colIdx1 == kA) ? 1.f : 0.f, (colIdx1 == kA + 1) ? 1.f : 0.f };
    v2f B2 = { (colIdx2 == kA) ? 1.f : 0.f, (colIdx2 == kA + 1) ? 1.f : 0.f };
    acc = __builtin_amdgcn_wmma_f32_16x16x4_f32(false, A1, false, B1, (short)0, acc, false, false);
    acc = __builtin_amdgcn_wmma_f32_16x16x4_f32(false, A2, false, B2, (short)0, acc, false, false);
  }

  const float4 a = ((const float4*)anchors)[i];
  const float by0 = pos ? acc[0] : a.x;
  const float bx0 = pos ? acc[1] : a.y;
  const float by1 = pos ? acc[2] : a.z;
  const float bx1 = pos ? acc[3] : a.w;

  // jax.random.uniform(kp, (B, N_AC)) element e, exact threefry counter pairing
  const uint32_t ue = (uint32_t)e;
  const uint32_t j  = (ue < HALF_TOTAL) ? ue : (ue - HALF_TOTAL);
  const uint32_t w  = tf_word(kp0, kp1, j, j + (uint32_t)HALF_TOTAL, ue >= HALF_TOTAL);
  const float rnd = __uint_as_float((w >> 9) | 0x3f800000u) - 1.0f;
  const float th  = NUM_POS / ((float)cnt[b] + 1e-6f);
  const float maskp = (basePos && (rnd < th)) ? 1.f : 0.f;

  // bbox2delta(bx_tgt_safe, anchor), clip +-10, * mask_pos
  const float wr  = a.w - a.y,  hr  = a.z - a.x;
  const float xcr = a.y + 0.5f * wr, ycr = a.x + 0.5f * hr;
  const float wrc = fmaxf(wr, EPS_F), hrc = fmaxf(hr, EPS_F);
  const float wl  = bx1 - bx0, hl = by1 - by0;
  const float xcl = bx0 + 0.5f * wl, ycl = by0 + 0.5f * hl;
  float d0 = (xcl - xcr) / wrc;
  float d1 = (ycl - ycr) / hrc;
  float d2 = logf(wl / wrc);
  float d3 = logf(hl / hrc);
  d0 = fminf(fmaxf(d0, -10.f), 10.f) * maskp;
  d1 = fminf(fmaxf(d1, -10.f), 10.f) * maskp;
  d2 = fminf(fmaxf(d2, -10.f), 10.f) * maskp;
  d3 = fminf(fmaxf(d3, -10.f), 10.f) * maskp;
  ((float4*)out)[e * 2 + 1] = make_float4(d0, d1, d2, d3);
}

// ---------------- Host side ----------------
static void host_threefry(uint32_t k0, uint32_t k1, uint32_t c0, uint32_t c1,
                          uint32_t* o0, uint32_t* o1) {
  uint32_t ks2 = k0 ^ k1 ^ 0x1BD11BDAu;
  uint32_t x0 = c0 + k0, x1 = c1 + k1;
#define TFR(rr) { x0 += x1; x1 = (x1 << rr) | (x1 >> (32 - rr)); x1 ^= x0; }
  TFR(13) TFR(15) TFR(26) TFR(6)   x0 += k1;  x1 += ks2 + 1u;
  TFR(17) TFR(29) TFR(16) TFR(24)  x0 += ks2; x1 += k0  + 2u;
  TFR(13) TFR(15) TFR(26) TFR(6)   x0 += k0;  x1 += k1  + 3u;
  TFR(17) TFR(29) TFR(16) TFR(24)  x0 += k1;  x1 += ks2 + 4u;
  TFR(13) TFR(15) TFR(26) TFR(6)   x0 += ks2; x1 += k0  + 5u;
#undef TFR
  *o0 = x0; *o1 = x1;
}

extern "C" void kernel_launch(void* const* d_in, const int* in_sizes, int n_in,
                              void* d_out, int out_size, void* d_ws, size_t ws_size,
                              hipStream_t stream) {
  const float* anchors = (const float*)d_in[0];   // (N_AC, 4) f32
  const float* gt      = (const float*)d_in[1];   // (B, 64, 4) f32
  const float* deltas  = (const float*)d_in[2];   // (B, N_AC, 4) f32
  float* out = (float*)d_out;                     // (B, N_AC, 8) f32
  int*   cnt = (int*)d_ws;                        // 8 per-batch pos counts

  // kp = jax.random.split(jax.random.key(42))[0]  (key data = [0, 42])
  uint32_t a0, a1, b0, b1;
  host_threefry(0u, 42u, 0u, 2u, &a0, &a1);
  host_threefry(0u, 42u, 1u, 3u, &b0, &b1);
  const uint32_t kp0 = a0, kp1 = b0;              // kn = (a1, b1) is dead code in ref

  rpn_init_counts<<<1, 32, 0, stream>>>(cnt);
  dim3 grid(N_AC / 256, B_SZ);
  rpn_assign<<<grid, 256, 0, stream>>>(anchors, gt, deltas, out, cnt);
  rpn_finalize<<<grid, 256, 0, stream>>>(anchors, gt, out, cnt, kp0, kp1);
}